// HyperbolicRGCNBlockLayer_38190849196694
// MI455X (gfx1250) — compile-verified
//
#include <hip/hip_runtime.h>
#include <math.h>

#define N_NODES   50000
#define N_EDGES   600000
#define FEAT      128
#define NUM_BASES 16   // sub-blocks of 8x8

typedef __attribute__((ext_vector_type(2))) float v2f;
typedef __attribute__((ext_vector_type(8))) float v8f;

__device__ __forceinline__ float wave_reduce_sum(float v) {
#pragma unroll
  for (int m = 16; m >= 1; m >>= 1) v += __shfl_xor(v, m, 32);
  return v;
}

// reduction within a 16-lane half (rows live per-half in the WMMA C layout)
__device__ __forceinline__ float half_reduce_sum(float v) {
#pragma unroll
  for (int m = 8; m >= 1; m >>= 1) v += __shfl_xor(v, m, 32);
  return v;
}

__device__ __forceinline__ float artanh_c(float x) {
  x = fminf(fmaxf(x, -1.0f + 1e-7f), 1.0f - 1e-7f);
  return 0.5f * logf((1.0f + x) / (1.0f - x));
}

// ---------------------------------------------------------------------------
// Kernel 1: per-node logmap0(h) and clamped node norm.  One wave per node.
// ---------------------------------------------------------------------------
__global__ void k_lognorm(const float* __restrict__ h,
                          float* __restrict__ log_h,
                          float* __restrict__ hn) {
  int wave = (blockIdx.x * blockDim.x + threadIdx.x) >> 5;
  int lane = threadIdx.x & 31;
  if (wave >= N_NODES) return;
  const float4 v = ((const float4*)(h + (size_t)wave * FEAT))[lane];
  float s = v.x * v.x + v.y * v.y + v.z * v.z + v.w * v.w;
  s = wave_reduce_sum(s);
  float n  = fmaxf(sqrtf(s), 1e-7f);         // _norm clip
  float sc = artanh_c(n) / n;                // sk = 1 for K = -1
  float4 o;
  o.x = v.x * sc; o.y = v.y * sc; o.z = v.z * sc; o.w = v.w * sc;
  ((float4*)(log_h + (size_t)wave * FEAT))[lane] = o;
  if (lane == 0) hn[wave] = n;
}

// ---------------------------------------------------------------------------
// Kernel 2: mx = h @ loop_weight.T via V_WMMA_F32_16X16X4_F32.
// One wave per 16-row tile, sweeping all 8 column tiles (full N=128).
// mx[m,n] = sum_k h[m,k] * loop_weight[n*128 + k]
// ---------------------------------------------------------------------------
__global__ void k_loop_gemm(const float* __restrict__ h,
                            const float* __restrict__ lw,
                            float* __restrict__ mx) {
  const int MT = N_NODES / 16;  // 3125 row tiles
  int wave = (blockIdx.x * blockDim.x + threadIdx.x) >> 5;
  int lane = threadIdx.x & 31;
  if (wave >= MT) return;
  const int m0   = wave * 16;
  const int half = lane >> 4;   // 0: K pair {0,1}, 1: K pair {2,3}
  const int l16  = lane & 15;

  v8f c[8] = {};
  const float* arow = h + (size_t)(m0 + l16) * FEAT + 2 * half;

#pragma unroll 2
  for (int k0 = 0; k0 < FEAT; k0 += 4) {
    v2f a;
    a.x = arow[k0];
    a.y = arow[k0 + 1];
#pragma unroll
    for (int t = 0; t < 8; ++t) {
      const float* bp = lw + (size_t)(t * 16 + l16) * FEAT + k0 + 2 * half;
      v2f b;
      b.x = bp[0];
      b.y = bp[1];
      c[t] = __builtin_amdgcn_wmma_f32_16x16x4_f32(
          false, a, false, b, (short)0, c[t], false, false);
    }
  }

  // C/D layout: VGPR r holds row m0 + r + 8*half, column l16 (+16*t)
#pragma unroll
  for (int t = 0; t < 8; ++t) {
#pragma unroll
    for (int r = 0; r < 8; ++r) {
      int row = m0 + r + 8 * half;
      mx[(size_t)row * FEAT + t * 16 + l16] = c[t][r];
    }
  }
}

// ---------------------------------------------------------------------------
// Kernel 3: per-edge block-diagonal transform + expmap0 + atomic scatter.
// One wave per edge. Lane L: base b = L>>1, outputs o = b*8 + (L&1)*4 .. +3.
// ---------------------------------------------------------------------------
__global__ void k_edges(const float* __restrict__ log_h,
                        const float* __restrict__ weight,
                        const int* __restrict__ src,
                        const int* __restrict__ dst,
                        const int* __restrict__ etype,
                        float* __restrict__ agg) {
  int wave = (blockIdx.x * blockDim.x + threadIdx.x) >> 5;
  int lane = threadIdx.x & 31;
  if (wave >= N_EDGES) return;
  const int s  = src[wave];
  const int d  = dst[wave];
  const int et = etype[wave];

  const int b  = lane >> 1;          // base index 0..15
  const int oq = (lane & 1) * 4;     // output quad within base

  const float* xl = log_h + (size_t)s * FEAT + b * 8;
  const float* wp = weight + (size_t)et * (NUM_BASES * 64) + b * 64 + oq;

  float a0 = 0.f, a1 = 0.f, a2 = 0.f, a3 = 0.f;
#pragma unroll
  for (int i = 0; i < 8; ++i) {
    const float  xv = xl[i];
    const float4 wv = *(const float4*)(wp + i * 8);
    a0 = fmaf(xv, wv.x, a0);
    a1 = fmaf(xv, wv.y, a1);
    a2 = fmaf(xv, wv.z, a2);
    a3 = fmaf(xv, wv.w, a3);
  }

  // expmap0: tanh(n) * u / n, n = clamp(||u||, 1e-7)
  float ss = a0 * a0 + a1 * a1 + a2 * a2 + a3 * a3;
  ss = wave_reduce_sum(ss);
  float n  = fmaxf(sqrtf(ss), 1e-7f);
  float sc = tanhf(n) / n;

  float* op = agg + (size_t)d * FEAT + b * 8 + oq;
  unsafeAtomicAdd(op + 0, a0 * sc);
  unsafeAtomicAdd(op + 1, a1 * sc);
  unsafeAtomicAdd(op + 2, a2 * sc);
  unsafeAtomicAdd(op + 3, a3 * sc);
}

// ---------------------------------------------------------------------------
// Kernel 4: per-node combine.  loop_msg = mobius_matvec nonlinearity on mx,
// x = agg*norm, out = mobius_add(x, loop_msg, k=-1).  One wave per node.
// ---------------------------------------------------------------------------
__global__ void k_combine(const float* __restrict__ agg,
                          const float* __restrict__ mx,
                          const float* __restrict__ hn,
                          const float* __restrict__ nodenorm,
                          float* __restrict__ out) {
  int wave = (blockIdx.x * blockDim.x + threadIdx.x) >> 5;
  int lane = threadIdx.x & 31;
  if (wave >= N_NODES) return;

  const float4 mv = ((const float4*)(mx + (size_t)wave * FEAT))[lane];
  float mxn2 = wave_reduce_sum(mv.x * mv.x + mv.y * mv.y + mv.z * mv.z + mv.w * mv.w);

  const float xn  = hn[wave];                   // clamped ||h||
  const float mxn = fmaxf(sqrtf(mxn2), 1e-7f);
  const float lsc = (mxn2 == 0.0f) ? 0.0f
                                   : tanhf(mxn / xn * artanh_c(xn)) / mxn;
  float4 y;
  y.x = mv.x * lsc; y.y = mv.y * lsc; y.z = mv.z * lsc; y.w = mv.w * lsc;

  const float nn = nodenorm[wave];
  const float4 av = ((const float4*)(agg + (size_t)wave * FEAT))[lane];
  float4 x;
  x.x = av.x * nn; x.y = av.y * nn; x.z = av.z * nn; x.w = av.w * nn;

  float x2 = wave_reduce_sum(x.x * x.x + x.y * x.y + x.z * x.z + x.w * x.w);
  float y2 = wave_reduce_sum(y.x * y.x + y.y * y.y + y.z * y.z + y.w * y.w);
  float xy = wave_reduce_sum(x.x * y.x + x.y * y.y + x.z * y.z + x.w * y.w);

  const float k  = -1.0f;
  const float ca = 1.0f - 2.0f * k * xy - k * y2;       // coeff on x
  const float cb = 1.0f + k * x2;                       // coeff on y
  float den = 1.0f - 2.0f * k * xy + k * k * x2 * y2;
  den = fmaxf(den, 1e-15f);
  const float inv = 1.0f / den;

  float4 o;
  o.x = (ca * x.x + cb * y.x) * inv;
  o.y = (ca * x.y + cb * y.y) * inv;
  o.z = (ca * x.z + cb * y.z) * inv;
  o.w = (ca * x.w + cb * y.w) * inv;
  ((float4*)(out + (size_t)wave * FEAT))[lane] = o;
}

// ---------------------------------------------------------------------------
extern "C" void kernel_launch(void* const* d_in, const int* in_sizes, int n_in,
                              void* d_out, int out_size, void* d_ws, size_t ws_size,
                              hipStream_t stream) {
  const float* h        = (const float*)d_in[0];
  const float* weight   = (const float*)d_in[1];
  const float* lw       = (const float*)d_in[2];
  const float* nodenorm = (const float*)d_in[3];
  const int*   src      = (const int*)d_in[4];
  const int*   dst      = (const int*)d_in[5];
  const int*   etype    = (const int*)d_in[6];
  float*       out      = (float*)d_out;

  float* ws    = (float*)d_ws;
  const size_t NF = (size_t)N_NODES * FEAT;
  float* log_h = ws;
  float* mx    = ws + NF;
  float* agg   = ws + 2 * NF;
  float* hn    = ws + 3 * NF;

  hipMemsetAsync(agg, 0, NF * sizeof(float), stream);

  {  // log map + norms: one wave per node
    int waves = N_NODES, threads = waves * 32;
    k_lognorm<<<(threads + 255) / 256, 256, 0, stream>>>(h, log_h, hn);
  }
  {  // WMMA GEMM: one wave per 16-row tile
    int waves = N_NODES / 16, threads = waves * 32;
    k_loop_gemm<<<(threads + 255) / 256, 256, 0, stream>>>(h, lw, mx);
  }
  {  // edges: one wave per edge
    long long threads = (long long)N_EDGES * 32;
    k_edges<<<(int)((threads + 255) / 256), 256, 0, stream>>>(log_h, weight, src,
                                                             dst, etype, agg);
  }
  {  // combine
    int waves = N_NODES, threads = waves * 32;
    k_combine<<<(threads + 255) / 256, 256, 0, stream>>>(agg, mx, hn, nodenorm, out);
  }
}